// Projector_59236188947235
// MI455X (gfx1250) — compile-verified
//
#include <hip/hip_runtime.h>

#define SIDE  128
#define BATCH 16

// Padded volume: z,y get 1 zero-plane each side; x gets 2 zero-columns each side
// (so the paired 8-byte load reads genuine zeros even when both x-corners are OOB).
#define PZ 130
#define PY 130
#define PX 132
#define PAD_ELEMS (PZ * PY * PX)

// 2-float vector loadable from a merely 4-byte-aligned address.
typedef float vfloat2 __attribute__((ext_vector_type(2)));

__device__ __forceinline__ vfloat2 load_pair(const float* __restrict__ p) {
    vfloat2 v;
    __builtin_memcpy(&v, p, sizeof(v));   // dword-aligned b64 load (DWORD mode ignores addr[1:0])
    return v;
}

// ---------------- pad pass: d_ws <- zero-padded volume ----------------
// One block per padded (z,y) row; threads walk x. (z,y) from blockIdx => SALU-only decomposition.
__global__ __launch_bounds__(160)
void Projector_59236188947235_pad(const float* __restrict__ vol,
                                  float* __restrict__ pvol) {
    const int zy = blockIdx.x;          // 0 .. PZ*PY-1 (uniform -> scalar div/mod)
    const int z  = zy / PY;
    const int y  = zy % PY;
    const int x  = threadIdx.x;
    if (x >= PX) return;

    float v = 0.0f;
    if (x >= 2 && x < 2 + SIDE && y >= 1 && y < 1 + SIDE && z >= 1 && z < 1 + SIDE) {
        v = vol[((z - 1) * SIDE + (y - 1)) * SIDE + (x - 2)];
    }
    pvol[zy * PX + x] = v;
}

// ---------------- fast path: mask-free trilinear march over padded volume ----------------
__global__ __launch_bounds__(256)
void Projector_59236188947235_kernel(const float* __restrict__ pvol,
                                     const float* __restrict__ rot,
                                     float* __restrict__ out) {
    const int b = blockIdx.x >> 6;                                   // uniform -> SGPR
    const int i = ((blockIdx.x & 63) << 1) | (threadIdx.x >> 7);
    const int j = threadIdx.x & (SIDE - 1);

    const float* __restrict__ R = rot + b * 9;                       // uniform -> s_load
    const float r00 = R[0], r01 = R[1], r02 = R[2];
    const float r10 = R[3], r11 = R[4], r12 = R[5];
    const float r20 = R[6], r21 = R[7], r22 = R[8];

    const float step  = 2.0f / (SIDE - 1);
    const float scale = 0.5f * (SIDE - 1);          // 63.5
    const float cy = fmaf(step, (float)j, -1.0f);   // coords = (lin[j], lin[i], lin[k])
    const float cx = fmaf(step, (float)i, -1.0f);

    // ix(k) = ((cy*r00 + cx*r10 + cz*r20) + 1) * scale + pad_offset
    // pad offsets baked into the affine base: x:+2, y:+1, z:+1 (floor commutes up to 1 ulp;
    // trilinear is continuous across cell boundaries so this is tolerance-safe).
    const float dix = r20 * scale, diy = r21 * scale, diz = r22 * scale;  // SGPR
    const float tix = fmaf(fmaf(cx, r10, cy * r00), scale, scale) + 2.0f;
    const float tiy = fmaf(fmaf(cx, r11, cy * r01), scale, scale) + 1.0f;
    const float tiz = fmaf(fmaf(cx, r12, cy * r02), scale, scale) + 1.0f;

    float acc = 0.0f;

    #pragma unroll 4
    for (int k = 0; k < SIDE; ++k) {
        const float cz = fmaf(step, (float)k, -1.0f);
        const float ix = fmaf(cz, dix, tix);        // already in padded index space
        const float iy = fmaf(cz, diy, tiy);
        const float iz = fmaf(cz, diz, tiz);

        const float fx = floorf(ix), fy = floorf(iy), fz = floorf(iz);
        const int   x0 = (int)fx,    y0 = (int)fy,    z0 = (int)fz;
        const float wx = ix - fx,    wy = iy - fy,    wz = iz - fz;

        // Clamp into padded space (v_med3_i32). Border voxels are zero, which
        // reproduces the reference's per-corner zero-padding with no weight masks.
        const int pxb = min(max(x0, 0), PX - 2);    // pair base: (pxb, pxb+1)
        const int py0 = min(max(y0, 0), PY - 1);
        const int py1 = min(max(y0 + 1, 0), PY - 1);
        const int pz0 = min(max(z0, 0), PZ - 1);
        const int pz1 = min(max(z0 + 1, 0), PZ - 1);

        const int r00i = (pz0 * PY + py0) * PX + pxb;
        const int r01i = (pz0 * PY + py1) * PX + pxb;
        const int r10i = (pz1 * PY + py0) * PX + pxb;
        const int r11i = (pz1 * PY + py1) * PX + pxb;

        const vfloat2 p00 = load_pair(pvol + r00i);
        const vfloat2 p01 = load_pair(pvol + r01i);
        const vfloat2 p10 = load_pair(pvol + r10i);
        const vfloat2 p11 = load_pair(pvol + r11i);

        // Lerp form: a + w*(b-a)
        const float l00 = fmaf(wx, p00.y - p00.x, p00.x);
        const float l01 = fmaf(wx, p01.y - p01.x, p01.x);
        const float l10 = fmaf(wx, p10.y - p10.x, p10.x);
        const float l11 = fmaf(wx, p11.y - p11.x, p11.x);

        const float q0 = fmaf(wy, l01 - l00, l00);
        const float q1 = fmaf(wy, l11 - l10, l10);

        acc += fmaf(wz, q1 - q0, q0);
    }

    __builtin_nontemporal_store(acc, out + (b * SIDE + i) * SIDE + j);
}

// ---------------- fallback (ws too small): masked gather, no padded copy ----------------
__global__ __launch_bounds__(256)
void Projector_59236188947235_masked(const float* __restrict__ vol,
                                     const float* __restrict__ rot,
                                     float* __restrict__ out) {
    const int b = blockIdx.x >> 6;
    const int i = ((blockIdx.x & 63) << 1) | (threadIdx.x >> 7);
    const int j = threadIdx.x & (SIDE - 1);

    const float* __restrict__ R = rot + b * 9;
    const float r00 = R[0], r01 = R[1], r02 = R[2];
    const float r10 = R[3], r11 = R[4], r12 = R[5];
    const float r20 = R[6], r21 = R[7], r22 = R[8];

    const float step  = 2.0f / (SIDE - 1);
    const float scale = 0.5f * (SIDE - 1);
    const float cy = fmaf(step, (float)j, -1.0f);
    const float cx = fmaf(step, (float)i, -1.0f);
    const float dix = r20 * scale, diy = r21 * scale, diz = r22 * scale;
    const float tix = fmaf(fmaf(cx, r10, cy * r00), scale, scale);
    const float tiy = fmaf(fmaf(cx, r11, cy * r01), scale, scale);
    const float tiz = fmaf(fmaf(cx, r12, cy * r02), scale, scale);

    float acc = 0.0f;
    #pragma unroll 2
    for (int k = 0; k < SIDE; ++k) {
        const float cz = fmaf(step, (float)k, -1.0f);
        const float ix = fmaf(cz, dix, tix);
        const float iy = fmaf(cz, diy, tiy);
        const float iz = fmaf(cz, diz, tiz);
        const float fx = floorf(ix), fy = floorf(iy), fz = floorf(iz);
        const int   x0 = (int)fx,    y0 = (int)fy,    z0 = (int)fz;
        const float wx = ix - fx,    wy = iy - fy,    wz = iz - fz;

        const float wx0 = (x0 >= 0  && x0 <= SIDE - 1) ? (1.0f - wx) : 0.0f;
        const float wx1 = (x0 >= -1 && x0 <= SIDE - 2) ? wx          : 0.0f;
        const float wy0 = (y0 >= 0  && y0 <= SIDE - 1) ? (1.0f - wy) : 0.0f;
        const float wy1 = (y0 >= -1 && y0 <= SIDE - 2) ? wy          : 0.0f;
        const float wz0 = (z0 >= 0  && z0 <= SIDE - 1) ? (1.0f - wz) : 0.0f;
        const float wz1 = (z0 >= -1 && z0 <= SIDE - 2) ? wz          : 0.0f;

        const int yc0 = min(max(y0,     0), SIDE - 1);
        const int yc1 = min(max(y0 + 1, 0), SIDE - 1);
        const int zc0 = min(max(z0,     0), SIDE - 1);
        const int zc1 = min(max(z0 + 1, 0), SIDE - 1);
        const int xb  = min(max(x0, 0), SIDE - 2);
        const bool xflip = (x0 != xb);

        const vfloat2 p00 = load_pair(vol + (zc0 * SIDE + yc0) * SIDE + xb);
        const vfloat2 p01 = load_pair(vol + (zc0 * SIDE + yc1) * SIDE + xb);
        const vfloat2 p10 = load_pair(vol + (zc1 * SIDE + yc0) * SIDE + xb);
        const vfloat2 p11 = load_pair(vol + (zc1 * SIDE + yc1) * SIDE + xb);

        const float a00 = xflip ? p00.y : p00.x, b00 = xflip ? p00.x : p00.y;
        const float a01 = xflip ? p01.y : p01.x, b01 = xflip ? p01.x : p01.y;
        const float a10 = xflip ? p10.y : p10.x, b10 = xflip ? p10.x : p10.y;
        const float a11 = xflip ? p11.y : p11.x, b11 = xflip ? p11.x : p11.y;

        const float l00 = fmaf(wx0, a00, wx1 * b00);
        const float l01 = fmaf(wx0, a01, wx1 * b01);
        const float l10 = fmaf(wx0, a10, wx1 * b10);
        const float l11 = fmaf(wx0, a11, wx1 * b11);
        const float q0  = fmaf(wy0, l00, wy1 * l01);
        const float q1  = fmaf(wy0, l10, wy1 * l11);
        acc = fmaf(wz0, q0, fmaf(wz1, q1, acc));
    }
    __builtin_nontemporal_store(acc, out + (b * SIDE + i) * SIDE + j);
}

extern "C" void kernel_launch(void* const* d_in, const int* in_sizes, int n_in,
                              void* d_out, int out_size, void* d_ws, size_t ws_size,
                              hipStream_t stream) {
    const float* vol = (const float*)d_in[0];   // [128,128,128] f32
    const float* rot = (const float*)d_in[1];   // [16,3,3] f32
    float* out = (float*)d_out;                 // [16,1,128,128] f32

    const int blocks = BATCH * (SIDE * SIDE / 256);   // 1024

    if (ws_size >= (size_t)PAD_ELEMS * sizeof(float)) {
        float* pvol = (float*)d_ws;
        Projector_59236188947235_pad<<<dim3(PZ * PY), dim3(160), 0, stream>>>(vol, pvol);
        Projector_59236188947235_kernel<<<dim3(blocks), dim3(256), 0, stream>>>(pvol, rot, out);
    } else {
        Projector_59236188947235_masked<<<dim3(blocks), dim3(256), 0, stream>>>(vol, rot, out);
    }
}